// ROIPool_7945689498106
// MI455X (gfx1250) — compile-verified
//
#include <hip/hip_runtime.h>

// Problem constants (fixed by setup_inputs)
#define NIMG  4
#define CCH   256
#define HH    200
#define WW    272
#define O     7
#define MAXCK 3             // max b128 chunks per row: ceil((271+3+1)/128)
#define BUFF  (MAXCK * 128) // 384 floats per row buffer (16B-aligned granules)
#define NBUF  3             // pipeline depth (rows in flight)

// ---------------------------------------------------------------------------
// Async global -> LDS copies (CDNA5 gfx1250). Each enabled lane copies 16B
// from (SGPR64 base + VGPR32 byte offset) into LDS at the wave-relative byte
// address in the first VGPR operand. Tracked by ASYNCcnt (in-order complete).
// ---------------------------------------------------------------------------
__device__ __forceinline__ void async_ld_b128(unsigned lds_byte_off,
                                              unsigned long long gbase,
                                              unsigned gbyte_off) {
  asm volatile("global_load_async_to_lds_b128 %0, %1, %2"
               :
               : "v"(lds_byte_off), "v"(gbyte_off), "s"(gbase)
               : "memory");
}

// Wait until ASYNCcnt <= n (block-uniform n in 0..6; immediates only)
__device__ __forceinline__ void wait_async_le(int n) {
  switch (n) {
    case 0:  asm volatile("s_wait_asynccnt 0" ::: "memory"); break;
    case 1:  asm volatile("s_wait_asynccnt 1" ::: "memory"); break;
    case 2:  asm volatile("s_wait_asynccnt 2" ::: "memory"); break;
    case 3:  asm volatile("s_wait_asynccnt 3" ::: "memory"); break;
    case 4:  asm volatile("s_wait_asynccnt 4" ::: "memory"); break;
    case 5:  asm volatile("s_wait_asynccnt 5" ::: "memory"); break;
    default: asm volatile("s_wait_asynccnt 6" ::: "memory"); break;
  }
}

// Drain this wave's LDS reads so a row buffer can be re-targeted by the async
// engine (async LDS writes are unordered vs this wave's in-flight ds_loads).
__device__ __forceinline__ void wait_dscnt0() {
  asm volatile("s_wait_dscnt 0" ::: "memory");
}

// ---------------------------------------------------------------------------
// One block per (roi, oy). 8 waves; wave w handles channels w, w+8, ...
// Depth-3 pipeline over the flattened (channel, row) stream with B128 async
// staging of the 16B-aligned column window [xa, xa+span).
// ---------------------------------------------------------------------------
extern "C" __global__ __launch_bounds__(256)
void roi_pool_kernel(const float* __restrict__ fm,
                     const int* __restrict__ rois,
                     float* __restrict__ out) {
  alignas(16) __shared__ float buf[8][NBUF][BUFF];  // per-wave triple buffer

  const int r    = blockIdx.x;
  const int oy   = blockIdx.y;
  const int lane = threadIdx.x & 31;
  const int wave = threadIdx.x >> 5;

  // ROI (uniform per block -> scalar loads)
  const int n    = rois[r * 5 + 0];
  const int x1   = rois[r * 5 + 1];
  const int y1   = rois[r * 5 + 2];
  const int x2   = rois[r * 5 + 3];
  const int y2   = rois[r * 5 + 4];
  const int w_in = x2 - x1 + 1;
  const int h_in = y2 - y1 + 1;

  // Aligned staging window: columns [xa, xa+span), 16B-aligned global start.
  const int xa    = x1 & ~3;
  const int shift = x1 - xa;           // 0..3 front pad
  const int span  = x2 - xa + 1;       // staged columns; shift + w_in == span
  const int nck   = (span + 127) >> 7; // b128 issues per row (uniform, 1..3)

  // Row range of this oy bin: [hs, he)  (floor start, ceil end)
  const int hs = y1 + (oy * h_in) / O;
  const int he = y1 + ((oy + 1) * h_in + (O - 1)) / O;
  const int bh = he - hs;

  // x-bin boundaries (uniform; fully unrolled use keeps them in registers)
  int S[O], E[O];
#pragma unroll
  for (int b = 0; b < O; ++b) {
    S[b] = (b * w_in) / O;
    E[b] = ((b + 1) * w_in + (O - 1)) / O;
  }

  unsigned ldsb[NBUF];
#pragma unroll
  for (int i = 0; i < NBUF; ++i)
    ldsb[i] = (unsigned)(size_t)(&buf[wave][i][0]);
  const unsigned long long gbase = (unsigned long long)(size_t)fm;

  // Issue cursor over the flattened (channel, row) stream for this wave.
  int    ic     = wave;
  int    iy     = hs;
  size_t iplane = ((size_t)(n * CCH + ic)) * ((size_t)HH * WW);
  const size_t planestep = (size_t)8 * HH * WW;

  auto issue_row = [&](unsigned lbase) {
    const size_t rowelem = iplane + (size_t)iy * WW + (size_t)xa; // %4 == 0
#pragma unroll
    for (int k = 0; k < MAXCK; ++k) {
      const int u = (k << 7) + (lane << 2);   // granule start col rel. xa
      if (k < nck && u < span) {
        async_ld_b128(lbase + 4u * (unsigned)u, gbase,
                      (unsigned)(4u * (rowelem + (size_t)u)));
      }
    }
    if (++iy == he) { iy = hs; ic += 8; iplane += planestep; }
  };

  // Prologue: two rows in flight (stream always has >= 32 rows).
  issue_row(ldsb[0]);
  issue_row(ldsb[1]);
  int pending = 2;  // rows issued but not yet consumed
  int p = 0;        // buffer of the next row to consume

  for (int c = wave; c < CCH; c += 8) {
    // Per-lane column sums (float4 per chunk); pad columns accumulate
    // harmless garbage that the binning stage excludes by range.
    float4 cs[MAXCK];
#pragma unroll
    for (int k = 0; k < MAXCK; ++k) cs[k] = make_float4(0.f, 0.f, 0.f, 0.f);

    for (int y = hs; y < he; ++y) {
      if (ic < CCH) {
        int q = p + (NBUF - 1); if (q >= NBUF) q -= NBUF;
        issue_row(ldsb[q]);           // keep two rows ahead in flight
        ++pending;
      }
      wait_async_le((pending - 1) * nck);  // oldest row now complete

      const float4* s4 = reinterpret_cast<const float4*>(&buf[wave][p][0]);
#pragma unroll
      for (int k = 0; k < MAXCK; ++k) {
        if (k < nck) {                // uniform guard: skip unused chunks
          const float4 v = s4[(k << 5) + lane];
          cs[k].x += v.x; cs[k].y += v.y; cs[k].z += v.z; cs[k].w += v.w;
        }
      }
      wait_dscnt0();                  // buffer p may now be re-targeted
      --pending;
      if (++p == NBUF) p = 0;
    }

    // Bin the column sums into 7 x-bins (range tests exclude pad columns;
    // bins may overlap by one column, handled by independent range tests).
    float a[O];
#pragma unroll
    for (int b = 0; b < O; ++b) a[b] = 0.0f;
#pragma unroll
    for (int k = 0; k < MAXCK; ++k) {
      if (k < nck) {
        const int u0 = (k << 7) + (lane << 2);
        const float vv[4] = {cs[k].x, cs[k].y, cs[k].z, cs[k].w};
#pragma unroll
        for (int j = 0; j < 4; ++j) {
          const int t = u0 + j - shift;   // column index within the ROI
          const float v = vv[j];
#pragma unroll
          for (int b = 0; b < O; ++b)
            if (t >= S[b] && t < E[b]) a[b] += v;
        }
      }
    }

    // wave32 all-reduce of the 7 bin partials
#pragma unroll
    for (int b = 0; b < O; ++b) {
#pragma unroll
      for (int m = 16; m >= 1; m >>= 1) a[b] += __shfl_xor(a[b], m, 32);
    }

    // lanes 0..6 write the 7 outputs of this (r, c, oy, :)
    if (lane < O) {
      float s = a[0];
#pragma unroll
      for (int b = 1; b < O; ++b) s = (lane == b) ? a[b] : s;
      const int Sl = (lane * w_in) / O;
      const int El = ((lane + 1) * w_in + (O - 1)) / O;
      const float area = (float)(bh * (El - Sl));
      const size_t idx =
          (((size_t)r * CCH + (size_t)c) * O + (size_t)oy) * O + (size_t)lane;
      out[idx] = s / area;
    }
  }
}

// First R floats of d_out: img_idx as float
extern "C" __global__ void roi_idx_kernel(const int* __restrict__ rois,
                                          float* __restrict__ out, int R) {
  const int r = blockIdx.x * blockDim.x + threadIdx.x;
  if (r < R) out[r] = (float)rois[r * 5];
}

extern "C" void kernel_launch(void* const* d_in, const int* in_sizes, int n_in,
                              void* d_out, int out_size, void* d_ws, size_t ws_size,
                              hipStream_t stream) {
  const float* fm   = (const float*)d_in[0];
  const int*   rois = (const int*)d_in[1];
  const int    R    = in_sizes[1] / 5;
  float*       out  = (float*)d_out;

  if (R <= 0) return;

  roi_idx_kernel<<<(R + 255) / 256, 256, 0, stream>>>(rois, out, R);

  dim3 grid((unsigned)R, O);
  roi_pool_kernel<<<grid, 256, 0, stream>>>(fm, rois, out + R);
}